// Decoder_42253888258823
// MI455X (gfx1250) — compile-verified
//
#include <hip/hip_runtime.h>
#include <math.h>

// Persistent GRU decoder: ONE kernel launch for all T=512 steps.
// Key insight: the recurrence is independent per batch row, and each block
// owns 16 rows exclusively, so no inter-block sync is ever needed.
//  - weights/biases resident in LDS for the whole kernel (~277KB of 320KB/WGP)
//  - hidden state for the block's 16 rows lives in LDS across all steps
//  - x tile double-buffered through registers (one full step of latency hiding)
//  - per step: gi/gh WMMA f32 16x16x4 -> gates -> h update -> out-proj WMMA
//              -> wave-shuffle log_softmax -> masked store

typedef __attribute__((ext_vector_type(2))) float v2f;
typedef __attribute__((ext_vector_type(8))) float v8f;

#define T_LEN 512
#define B_SZ  1024
#define V_SZ  64
#define H_SZ  100
#define N3H   300
#define NPAD  304
#define XLD   68   // padded row stride (floats) for W_ih / xbuf: bank-conflict-free

__device__ __forceinline__ v8f wmma4(v2f a, v2f b, v8f c) {
  // (neg_a, A, neg_b, B, c_mod, C, reuse_a, reuse_b)
  return __builtin_amdgcn_wmma_f32_16x16x4_f32(false, a, false, b, (short)0, c,
                                               false, false);
}

__global__ __launch_bounds__(512, 1) void gru_decode_kernel(
    const float* __restrict__ x,      // [T,B,V]
    const float* __restrict__ h0,     // [B,H]
    const int* __restrict__ lengths,  // [B]
    const float* __restrict__ W_ih,   // [300,64]
    const float* __restrict__ W_hh,   // [300,100]
    const float* __restrict__ b_ih,   // [300]
    const float* __restrict__ b_hh,   // [300]
    const float* __restrict__ W_out,  // [64,100]
    const float* __restrict__ b_out,  // [64]
    float* __restrict__ out) {        // [T,B,V]
  __shared__ float Wih_s[N3H * XLD];     // 81.6 KB
  __shared__ float Whh_s[N3H * H_SZ];    // 120 KB
  __shared__ float Wout_s[V_SZ * H_SZ];  // 25.6 KB
  __shared__ float bih_s[N3H], bhh_s[N3H], bout_s[V_SZ];
  __shared__ int   len_s[16];
  __shared__ float hbuf[16][H_SZ];       // running hidden state (block's rows)
  __shared__ float xbuf[16][XLD];        // current x tile
  __shared__ float gi_s[16][NPAD];
  __shared__ float gh_s[16][NPAD];
  __shared__ float lg[16][V_SZ];

  const int tid  = threadIdx.x;
  const int wave = tid >> 5;   // 0..15
  const int lane = tid & 31;
  const int hl   = lane >> 4;  // half-wave: K sub-columns / upper-8 M rows
  const int l15  = lane & 15;
  const int row0 = blockIdx.x * 16;

  // ---- one-time: stage weights, biases, lengths, initial hidden into LDS ---
  for (int i = tid; i < N3H * 16; i += 512) {  // W_ih: 16 float4 per row
    const int r = i >> 4, c = (i & 15) << 2;
    *(float4*)&Wih_s[r * XLD + c] = *(const float4*)&W_ih[r * V_SZ + c];
  }
  for (int i = tid; i < N3H * 25; i += 512) {  // W_hh: 25 float4 per row
    const int r = i / 25, c = (i - r * 25) << 2;
    *(float4*)&Whh_s[r * H_SZ + c] = *(const float4*)&W_hh[r * H_SZ + c];
  }
  for (int i = tid; i < V_SZ * 25; i += 512) {  // W_out
    const int r = i / 25, c = (i - r * 25) << 2;
    *(float4*)&Wout_s[r * H_SZ + c] = *(const float4*)&W_out[r * H_SZ + c];
  }
  for (int i = tid; i < N3H; i += 512) { bih_s[i] = b_ih[i]; bhh_s[i] = b_hh[i]; }
  if (tid < V_SZ) bout_s[tid] = b_out[tid];
  if (tid < 16) len_s[tid] = lengths[row0 + tid];
  for (int i = tid; i < 16 * H_SZ; i += 512) {
    const int m = i / H_SZ, j = i - m * H_SZ;
    hbuf[m][j] = h0[(size_t)(row0 + m) * H_SZ + j];
  }

  // ---- x tile register double-buffer: thread owns one float2 of 16x64 tile --
  const int xr = tid >> 5;         // row 0..15
  const int xc = (tid & 31) << 1;  // col 0,2,..,62
  v2f xreg = *(const v2f*)(x + (size_t)(row0 + xr) * V_SZ + xc);  // t = 0

#pragma unroll 1
  for (int t = 0; t < T_LEN; ++t) {
    // deposit current x tile, then immediately issue next step's global load
    *(v2f*)&xbuf[xr][xc] = xreg;
    if (t + 1 < T_LEN)
      xreg = *(const v2f*)(x + ((size_t)(t + 1) * B_SZ + row0 + xr) * V_SZ + xc);
    if (t + 8 < T_LEN)  // keep the x stream deep in L2 (global_prefetch_b8)
      __builtin_prefetch(x + ((size_t)(t + 8) * B_SZ + row0 + xr) * V_SZ + xc, 0, 1);
    __syncthreads();  // xbuf ready (also fences the one-time LDS init at t==0)

    // ---- Phase 1: gi (jobs 0..18, K=64) and gh (jobs 19..37, K=100) --------
    for (int job = wave; job < 38; job += 16) {
      v8f acc = {0.f, 0.f, 0.f, 0.f, 0.f, 0.f, 0.f, 0.f};
      if (job < 19) {
        const int nt = job;
        int n = nt * 16 + l15;
        if (n > N3H - 1) n = N3H - 1;  // clamp; padded cols never read
        const float* ap = &xbuf[l15][2 * hl];
        const float* bp = &Wih_s[n * XLD + 2 * hl];
#pragma unroll
        for (int kk = 0; kk < V_SZ; kk += 4)
          acc = wmma4(*(const v2f*)(ap + kk), *(const v2f*)(bp + kk), acc);
#pragma unroll
        for (int r = 0; r < 8; ++r)
          gi_s[r + 8 * hl][nt * 16 + l15] = acc[r];
      } else {
        const int nt = job - 19;
        int n = nt * 16 + l15;
        if (n > N3H - 1) n = N3H - 1;
        const float* ap = &hbuf[l15][2 * hl];
        const float* bp = &Whh_s[n * H_SZ + 2 * hl];
#pragma unroll
        for (int kk = 0; kk < H_SZ; kk += 4)
          acc = wmma4(*(const v2f*)(ap + kk), *(const v2f*)(bp + kk), acc);
#pragma unroll
        for (int r = 0; r < 8; ++r)
          gh_s[r + 8 * hl][nt * 16 + l15] = acc[r];
      }
    }
    __syncthreads();

    // ---- Phase 2: gates + masked hidden update (in-LDS) --------------------
    for (int idx = tid; idx < 16 * H_SZ; idx += 512) {
      const int m = idx / H_SZ;
      const int j = idx - m * H_SZ;
      const float gir = gi_s[m][j] + bih_s[j];
      const float ghr = gh_s[m][j] + bhh_s[j];
      const float giz = gi_s[m][H_SZ + j] + bih_s[H_SZ + j];
      const float ghz = gh_s[m][H_SZ + j] + bhh_s[H_SZ + j];
      const float gin = gi_s[m][2 * H_SZ + j] + bih_s[2 * H_SZ + j];
      const float ghn = gh_s[m][2 * H_SZ + j] + bhh_s[2 * H_SZ + j];
      const float r = 1.f / (1.f + expf(-(gir + ghr)));
      const float z = 1.f / (1.f + expf(-(giz + ghz)));
      const float n = tanhf(gin + r * ghn);
      const float hold = hbuf[m][j];
      const float hnew = (1.f - z) * n + z * hold;
      hbuf[m][j] = (t < len_s[m]) ? hnew : hold;
    }
    __syncthreads();

    // ---- Phase 3: logits = h_new @ W_out^T + b_out (4 tiles, K=100) --------
    if (wave < 4) {
      const int n = wave * 16 + l15;  // < 64
      v8f acc = {0.f, 0.f, 0.f, 0.f, 0.f, 0.f, 0.f, 0.f};
      const float* ap = &hbuf[l15][2 * hl];
      const float* bp = &Wout_s[n * H_SZ + 2 * hl];
#pragma unroll
      for (int kk = 0; kk < H_SZ; kk += 4)
        acc = wmma4(*(const v2f*)(ap + kk), *(const v2f*)(bp + kk), acc);
#pragma unroll
      for (int r = 0; r < 8; ++r)
        lg[r + 8 * hl][n] = acc[r] + bout_s[n];
    }
    __syncthreads();

    // ---- Phase 4: per-row log_softmax, masked store (wave w = row w) -------
    {
      const int m = wave;
      float v0 = lg[m][lane];
      float v1 = lg[m][lane + 32];
      float mx = fmaxf(v0, v1);
#pragma unroll
      for (int o = 16; o > 0; o >>= 1) mx = fmaxf(mx, __shfl_xor(mx, o, 32));
      float s = expf(v0 - mx) + expf(v1 - mx);
#pragma unroll
      for (int o = 16; o > 0; o >>= 1) s += __shfl_xor(s, o, 32);
      const float lse = mx + logf(s);
      const bool active = t < len_s[m];
      const size_t base = ((size_t)t * B_SZ + row0 + m) * V_SZ;
      out[base + lane]      = active ? (v0 - lse) : 0.f;
      out[base + lane + 32] = active ? (v1 - lse) : 0.f;
    }
    // next iteration's xbuf store is fenced from this step's phase-1 readers
    // by the post-phase-1 barrier; lg/gi_s/gh_s writers are likewise fenced.
  }
}

extern "C" void kernel_launch(void* const* d_in, const int* in_sizes, int n_in,
                              void* d_out, int out_size, void* d_ws,
                              size_t ws_size, hipStream_t stream) {
  (void)in_sizes; (void)n_in; (void)out_size; (void)d_ws; (void)ws_size;
  const float* x     = (const float*)d_in[0];
  const float* h0    = (const float*)d_in[1];
  const int*   lens  = (const int*)d_in[2];
  const float* W_ih  = (const float*)d_in[3];
  const float* W_hh  = (const float*)d_in[4];
  const float* b_ih  = (const float*)d_in[5];
  const float* b_hh  = (const float*)d_in[6];
  const float* W_out = (const float*)d_in[7];
  const float* b_out = (const float*)d_in[8];
  float* out = (float*)d_out;

  gru_decode_kernel<<<B_SZ / 16, 512, 0, stream>>>(
      x, h0, lens, W_ih, W_hh, b_ih, b_hh, W_out, b_out, out);
}